// LossSobolev_12773232739069
// MI455X (gfx1250) — compile-verified
//
#include <hip/hip_runtime.h>

typedef float v2f __attribute__((ext_vector_type(2)));
typedef float v8f __attribute__((ext_vector_type(8)));

#define B_  256
#define IN_ 512
#define H_  256
#define D_  16
#define M_  64

// ---------------------------------------------------------------------------
// Kernel A: fold the particle-mean into the weights.
//   W2bar[k,d] = (1/M) * sum_m W2[k, d*M + m]      (256 x 16)
//   b2bar[d]   = (1/M) * sum_m b2[d*M + m]         (16)
// W2 rows are contiguous, so each (k,d) reduction is 64 contiguous floats ->
// 16 x float4 loads (global_load_b128), perfectly coalesced across the wave.
// ---------------------------------------------------------------------------
__global__ void w2bar_kernel(const float* __restrict__ W2,
                             const float* __restrict__ b2,
                             float* __restrict__ w2bar,
                             float* __restrict__ b2bar) {
  int tid = blockIdx.x * blockDim.x + threadIdx.x;   // 0 .. 4095
  int k = tid >> 4;
  int d = tid & 15;
  const float4* p = (const float4*)(W2 + (size_t)k * (D_ * M_) + d * M_);
  float4 s4 = make_float4(0.f, 0.f, 0.f, 0.f);
#pragma unroll
  for (int i = 0; i < M_ / 4; ++i) {
    float4 v = p[i];
    s4.x += v.x; s4.y += v.y; s4.z += v.z; s4.w += v.w;
  }
  w2bar[k * D_ + d] = (s4.x + s4.y + s4.z + s4.w) * (1.0f / M_);

  if (blockIdx.x == 0 && threadIdx.x < D_) {
    float s = 0.f;
#pragma unroll
    for (int m = 0; m < M_; ++m) s += b2[threadIdx.x * M_ + m];
    b2bar[threadIdx.x] = s * (1.0f / M_);
  }
}

// ---------------------------------------------------------------------------
// Kernel B: h = tanh(x @ W1 + b1), 256x512x256 GEMM via V_WMMA_F32_16X16X4_F32.
// One wave per 16x16 output tile; 4 waves (128 threads) per block; 64 blocks.
// A fragment (16x4 f32, 2 VGPRs/lane): lanes 0-15 hold K=k0,k0+1 of row M=lane,
// lanes 16-31 hold K=k0+2,k0+3 of row M=lane-16.  B fragment mirrors this over
// N.  Accumulator: 8 VGPRs, VGPR r = rows {r, r+8} split across lane halves.
// ---------------------------------------------------------------------------
__global__ void gemm1_tanh_kernel(const float* __restrict__ x,
                                  const float* __restrict__ W1,
                                  const float* __restrict__ b1,
                                  float* __restrict__ h) {
  const int wave = threadIdx.x >> 5;
  const int lane = threadIdx.x & 31;
  const int tile = blockIdx.x * 4 + wave;     // 0..255
  const int tm = tile >> 4;                   // M tile
  const int tn = tile & 15;                   // N tile
  const int hi = lane >> 4;                   // lane half: 0 or 1
  const int l  = lane & 15;

  const float* arow = x  + (size_t)(tm * 16 + l) * IN_;  // row M = tm*16+l
  const float* bcol = W1 + (tn * 16 + l);                // col N = tn*16+l

  v8f acc = {};
#pragma unroll 4
  for (int k0 = 0; k0 < IN_; k0 += 4) {
    if ((k0 & 63) == 0 && k0 + 64 < IN_)
      __builtin_prefetch(arow + k0 + 64, 0, 0);          // global_prefetch_b8
    const int kb = k0 + hi * 2;
    v2f a; a.x = arow[kb];            a.y = arow[kb + 1];
    v2f b; b.x = bcol[(size_t)kb * H_]; b.y = bcol[(size_t)(kb + 1) * H_];
    acc = __builtin_amdgcn_wmma_f32_16x16x4_f32(
        /*neg_a=*/false, a, /*neg_b=*/false, b,
        /*c_mod=*/(short)0, acc, /*reuse_a=*/false, /*reuse_b=*/false);
  }

  const float bias = b1[tn * 16 + l];
#pragma unroll
  for (int r = 0; r < 8; ++r) {
    const int row = tm * 16 + r + hi * 8;
    h[(size_t)row * H_ + tn * 16 + l] = tanhf(acc[r] + bias);
  }
}

// ---------------------------------------------------------------------------
// Kernel C: mf = h @ W2bar + b2bar  (256x256x16 skinny GEMM, WMMA),
// then loss = 0.5/B * sum((y - mf)^2).  16 waves = 16 M-tiles x 1 N-tile in a
// single 512-thread block, so the final reduction needs no atomics and the
// scalar is written exactly once.
// ---------------------------------------------------------------------------
__global__ void gemm2_loss_kernel(const float* __restrict__ h,
                                  const float* __restrict__ w2bar,
                                  const float* __restrict__ b2bar,
                                  const float* __restrict__ y,
                                  float* __restrict__ out) {
  __shared__ float red[16];
  const int wave = threadIdx.x >> 5;   // M tile 0..15
  const int lane = threadIdx.x & 31;
  const int hi = lane >> 4;
  const int l  = lane & 15;            // N = d  (D_ == 16: one N tile)

  const float* arow = h + (size_t)(wave * 16 + l) * H_;
  const float* bcol = w2bar + l;

  v8f acc = {};
#pragma unroll 8
  for (int k0 = 0; k0 < H_; k0 += 4) {
    const int kb = k0 + hi * 2;
    v2f a; a.x = arow[kb];           a.y = arow[kb + 1];
    v2f b; b.x = bcol[kb * D_];      b.y = bcol[(kb + 1) * D_];
    acc = __builtin_amdgcn_wmma_f32_16x16x4_f32(
        false, a, false, b, (short)0, acc, false, false);
  }

  const float bias = b2bar[l];
  float s = 0.f;
#pragma unroll
  for (int r = 0; r < 8; ++r) {
    const int row = wave * 16 + r + hi * 8;
    const float mf = acc[r] + bias;
    const float diff = y[row * D_ + l] - mf;
    s += diff * diff;
  }

  // wave32 shuffle reduction, then cross-wave LDS reduction
  for (int off = 16; off > 0; off >>= 1) s += __shfl_down(s, off, 32);
  if (lane == 0) red[wave] = s;
  __syncthreads();
  if (threadIdx.x == 0) {
    float t = 0.f;
#pragma unroll
    for (int i = 0; i < 16; ++i) t += red[i];
    out[0] = (0.5f / (float)B_) * t;   // == mmd2 / M
  }
}

// ---------------------------------------------------------------------------
extern "C" void kernel_launch(void* const* d_in, const int* in_sizes, int n_in,
                              void* d_out, int out_size, void* d_ws, size_t ws_size,
                              hipStream_t stream) {
  const float* x  = (const float*)d_in[0];   // [256, 512]
  const float* y  = (const float*)d_in[1];   // [256, 16]
  const float* W1 = (const float*)d_in[2];   // [512, 256]
  const float* b1 = (const float*)d_in[3];   // [256]
  const float* W2 = (const float*)d_in[4];   // [256, 1024]
  const float* b2 = (const float*)d_in[5];   // [1024]

  float* ws    = (float*)d_ws;
  float* w2bar = ws;               // 4096 floats
  float* b2bar = ws + 4096;        // 16 floats (padded to 32)
  float* h     = ws + 4128;        // 65536 floats
  // total workspace: ~272 KB

  w2bar_kernel<<<dim3(16), dim3(256), 0, stream>>>(W2, b2, w2bar, b2bar);
  gemm1_tanh_kernel<<<dim3(64), dim3(128), 0, stream>>>(x, W1, b1, h);
  gemm2_loss_kernel<<<dim3(1), dim3(512), 0, stream>>>(h, w2bar, b2bar, y,
                                                       (float*)d_out);
}